// GCNLayer_13804024889768
// MI455X (gfx1250) — compile-verified
//
#include <hip/hip_runtime.h>

typedef float v2f __attribute__((ext_vector_type(2)));
typedef float v8f __attribute__((ext_vector_type(8)));

#define D_IN   64
#define D_CAT  128   // 2*D_IN
#define D_OUT  64

// ---------------------------------------------------------------------------
// Kernel 0: zero workspace (summed accumulator + degree)
// ---------------------------------------------------------------------------
__global__ void gcn_zero_kernel(float* __restrict__ p, long n) {
    long i = (long)blockIdx.x * blockDim.x + threadIdx.x;
    if (i < n) p[i] = 0.0f;
}

// ---------------------------------------------------------------------------
// Kernel 1: edge scatter.  16 lanes per edge; each lane moves 4 floats of the
// source row via float4 load + 4 global f32 atomics into summed[dst].
// h and summed both fit in the 192MB L2, so atomics resolve at L2 rates.
// ---------------------------------------------------------------------------
__global__ void gcn_scatter_kernel(const float* __restrict__ h,
                                   const int* __restrict__ esrc,
                                   const int* __restrict__ edst,
                                   float* __restrict__ summed,
                                   float* __restrict__ deg,
                                   int E) {
    int t    = blockIdx.x * blockDim.x + threadIdx.x;
    int e    = t >> 4;
    int lane = t & 15;
    if (e >= E) return;
    int s = esrc[e];
    int d = edst[e];
    const float4 v = *reinterpret_cast<const float4*>(h + (size_t)s * D_IN + lane * 4);
    float* o = summed + (size_t)d * D_IN + lane * 4;
    atomicAdd(o + 0, v.x);
    atomicAdd(o + 1, v.y);
    atomicAdd(o + 2, v.z);
    atomicAdd(o + 3, v.w);
    if (lane == 0) atomicAdd(deg + d, 1.0f);
}

// ---------------------------------------------------------------------------
// Kernel 2: deg -> 1/max(deg,1) in place
// ---------------------------------------------------------------------------
__global__ void gcn_invdeg_kernel(float* __restrict__ deg, int n) {
    int i = blockIdx.x * blockDim.x + threadIdx.x;
    if (i < n) deg[i] = 1.0f / fmaxf(deg[i], 1.0f);
}

// ---------------------------------------------------------------------------
// Kernel 3: fused concat+normalize+GEMM via V_WMMA_F32_16X16X4_F32.
// One wave computes a 16x64 output tile: 4 C accumulators (16x16 each),
// K-loop of 32 steps over the virtual 128-wide concat matrix:
//   k <  64 : A from h[m]         (h_dst half)
//   k >= 64 : A from summed[m] * invdeg[m]  (h_neigh half, scale fused in)
// W (64x128 f32 = 32KB) is staged in LDS per block; B fragments read from LDS.
// Bias folded into accumulator init.
// ---------------------------------------------------------------------------
__global__ void __launch_bounds__(128)
gcn_gemm_kernel(const float* __restrict__ h,
                const float* __restrict__ summed,
                const float* __restrict__ invdeg,
                const float* __restrict__ W,
                const float* __restrict__ bias,
                float* __restrict__ out,
                int M) {
    __shared__ float Wlds[D_OUT * D_CAT];  // 32 KB

    // cooperative W load: 8192 floats = 2048 float4, 128 threads
    {
        const float4* Wv = reinterpret_cast<const float4*>(W);
        float4*       Wl = reinterpret_cast<float4*>(Wlds);
        #pragma unroll
        for (int i = 0; i < (D_OUT * D_CAT / 4) / 128; ++i)
            Wl[threadIdx.x + i * 128] = Wv[threadIdx.x + i * 128];
    }
    __syncthreads();

    const int wave = threadIdx.x >> 5;
    const int lane = threadIdx.x & 31;
    const int m0   = (blockIdx.x * 4 + wave) * 16;
    if (m0 >= M) return;   // wave-uniform exit (EXEC stays all-ones for WMMA)

    const int half = lane >> 4;   // K-pair select for A/B fragments
    const int idx  = lane & 15;   // row (A) / column (B,C) within tile

    const float  invd = invdeg[m0 + idx];
    const float* arow = h      + (size_t)(m0 + idx) * D_IN;
    const float* srow = summed + (size_t)(m0 + idx) * D_IN;

    // C init = bias broadcast down each column (column of C == idx)
    v8f c[4];
    #pragma unroll
    for (int nt = 0; nt < 4; ++nt) {
        const float bb = bias[nt * 16 + idx];
        #pragma unroll
        for (int r = 0; r < 8; ++r) c[nt][r] = bb;
    }

    #pragma unroll
    for (int k0 = 0; k0 < D_CAT; k0 += 4) {
        const int kk = k0 + 2 * half;

        // A fragment: 16x4 f32, lane holds (row=idx, K=kk,kk+1)
        v2f a;
        if (k0 < D_IN) {
            a[0] = arow[kk];
            a[1] = arow[kk + 1];
        } else {
            a[0] = srow[kk - D_IN]     * invd;
            a[1] = srow[kk - D_IN + 1] * invd;
        }

        // 4 N-tiles reuse the same A fragment
        #pragma unroll
        for (int nt = 0; nt < 4; ++nt) {
            const float* wrow = Wlds + (nt * 16 + idx) * D_CAT + kk;
            v2f bf;
            bf[0] = wrow[0];
            bf[1] = wrow[1];
            c[nt] = __builtin_amdgcn_wmma_f32_16x16x4_f32(
                false, a, false, bf, (short)0, c[nt], false, false);
        }
    }

    // store: C element (r, idx) -> out[m0 + r + 8*half][nt*16 + idx]
    #pragma unroll
    for (int nt = 0; nt < 4; ++nt) {
        #pragma unroll
        for (int r = 0; r < 8; ++r) {
            const int m = m0 + r + 8 * half;
            out[(size_t)m * D_OUT + nt * 16 + idx] = c[nt][r];
        }
    }
}

// ---------------------------------------------------------------------------
extern "C" void kernel_launch(void* const* d_in, const int* in_sizes, int n_in,
                              void* d_out, int out_size, void* d_ws, size_t ws_size,
                              hipStream_t stream) {
    const float* h    = (const float*)d_in[0];
    const float* W    = (const float*)d_in[1];
    const float* bias = (const float*)d_in[2];
    const int*   esrc = (const int*)d_in[3];
    const int*   edst = (const int*)d_in[4];
    // d_in[5] = n_dst device scalar; recover on host instead (graph-capture safe)
    const int E     = in_sizes[3];
    const int n_dst = out_size / D_OUT;

    float* summed = (float*)d_ws;                      // n_dst * 64 floats
    float* deg    = summed + (size_t)n_dst * D_IN;     // n_dst floats
    float* out    = (float*)d_out;

    // zero accumulators (harness poisons ws; must re-init every call)
    const long zn = (long)n_dst * (D_IN + 1);
    gcn_zero_kernel<<<(unsigned)((zn + 255) / 256), 256, 0, stream>>>(summed, zn);

    // edge scatter: 16 threads per edge
    const long st = (long)E * 16;
    gcn_scatter_kernel<<<(unsigned)((st + 255) / 256), 256, 0, stream>>>(
        h, esrc, edst, summed, deg, E);

    gcn_invdeg_kernel<<<(n_dst + 255) / 256, 256, 0, stream>>>(deg, n_dst);

    // one wave per 16-row tile, 4 waves per block
    const int mtiles = (n_dst + 15) / 16;
    const int blocks = (mtiles + 3) / 4;
    gcn_gemm_kernel<<<blocks, 128, 0, stream>>>(h, summed, deg, W, bias, out, n_dst);
}